// LabelLoss_6030134084336
// MI455X (gfx1250) — compile-verified
//
#include <hip/hip_runtime.h>
#include <stdint.h>

// ---------------------------------------------------------------------------
// LabelLoss for MI455X (gfx1250).
// B=32, J=11, C=9, M=N=256.  Per (b,j): argmax over 64K-float heatmap slice,
// gather 9 pred channels at argmax, squared-error loss, sum over j.
//
// HBM-bound (~92 MB streamed => ~4us at 23.3 TB/s). No GEMM => no WMMA.
// CDNA5 path used: async global->LDS (global_load_async_to_lds_b128) with a
// double-buffered tile pipeline gated by s_wait_asynccnt (ASYNCcnt), wave32
// shuffle argmax reduction. Deterministic (no float atomics).
// ---------------------------------------------------------------------------

namespace {
constexpr int Bn = 32;
constexpr int Jn = 11;
constexpr int Cn = 9;
constexpr int Mdim = 256;
constexpr int Ndim = 256;
constexpr int MN = Mdim * Ndim;            // 65536 floats per (b,j) slice
constexpr int THREADS = 256;               // 8 wave32
constexpr int WAVES = THREADS / 32;        // 8
constexpr int TILE_F = 4096;               // floats per tile (16 KB)
constexpr int NTILES = MN / TILE_F;        // 16
constexpr int WAVE_F = TILE_F / WAVES;     // 512 floats per wave per tile
constexpr int KITER = WAVE_F / (32 * 4);   // 4 x b128 per lane per tile
} // namespace

// Async copy: 16 bytes per lane, global -> LDS, tracked by ASYNCcnt.
__device__ __forceinline__ void async_copy_b128(uint32_t lds_off, const void* gaddr) {
  asm volatile("global_load_async_to_lds_b128 %0, %1, off"
               :: "v"(lds_off), "v"(gaddr)
               : "memory");
}
__device__ __forceinline__ void wait_async_le4() {
  asm volatile("s_wait_asynccnt 4" ::: "memory");
}
__device__ __forceinline__ void wait_async_le0() {
  asm volatile("s_wait_asynccnt 0" ::: "memory");
}
__device__ __forceinline__ void wait_ds_0() {
  asm volatile("s_wait_dscnt 0" ::: "memory");
}

extern "C" __global__ void __launch_bounds__(THREADS)
labelloss_argmax_kernel(const float* __restrict__ pred,
                        const float* __restrict__ gt,
                        const float* __restrict__ hm,
                        float* __restrict__ ws) {
  __shared__ __align__(16) float tile[2][TILE_F];   // 32 KB double buffer
  __shared__ float sval[WAVES];
  __shared__ int   sidx[WAVES];

  const int bj   = blockIdx.x;          // 0 .. B*J-1
  const int b    = bj / Jn;
  const int tid  = threadIdx.x;
  const int wave = tid >> 5;
  const int lane = tid & 31;

  // Each wave owns a contiguous 2 KB chunk of each tile: it fills it with
  // async copies and later reads only that chunk back -> no barrier needed.
  const char* gbase =
      (const char*)(hm + (size_t)bj * MN) + (size_t)(wave * WAVE_F * 4 + lane * 16);
  const uint32_t lchunk = (uint32_t)(wave * WAVE_F * 4 + lane * 16);
  const uint32_t lbase[2] = {
      (uint32_t)(uintptr_t)(&tile[0][0]) + lchunk,   // addrspace(3) low 32 bits = LDS byte offset
      (uint32_t)(uintptr_t)(&tile[1][0]) + lchunk };

  auto issue_tile = [&](int t) {
    const char* g = gbase + (size_t)t * (TILE_F * 4);
    const uint32_t l = lbase[t & 1];
#pragma unroll
    for (int k = 0; k < KITER; ++k)          // each k: 32 lanes x 16B = 512B
      async_copy_b128(l + (uint32_t)(k * 512), g + k * 512);
  };

  float bestV = -__builtin_huge_valf();
  int   bestI = 0x7fffffff;

  issue_tile(0);
  issue_tile(1);

  for (int t = 0; t < NTILES; ++t) {
    if (t < NTILES - 1) wait_async_le4();    // tile t's 4 copies retired (in-order)
    else                wait_async_le0();

    const float* buf = &tile[t & 1][wave * WAVE_F];
#pragma unroll
    for (int k = 0; k < KITER; ++k) {
      const float4 v = *reinterpret_cast<const float4*>(buf + k * 128 + lane * 4);
      const int base = t * TILE_F + wave * WAVE_F + k * 128 + lane * 4;
      // ascending index scan: strict '>' keeps the first occurrence
      if (v.x > bestV) { bestV = v.x; bestI = base;     }
      if (v.y > bestV) { bestV = v.y; bestI = base + 1; }
      if (v.z > bestV) { bestV = v.z; bestI = base + 2; }
      if (v.w > bestV) { bestV = v.w; bestI = base + 3; }
    }

    if (t + 2 < NTILES) {
      wait_ds_0();          // our LDS reads of this buffer have retired
      issue_tile(t + 2);    // safe to overwrite buffer (t&1)
    }
  }

  // wave32 argmax reduction (tie -> smallest flat index, matching jnp.argmax)
#pragma unroll
  for (int off = 16; off > 0; off >>= 1) {
    const float ov = __shfl_xor(bestV, off, 32);
    const int   oi = __shfl_xor(bestI, off, 32);
    if (ov > bestV || (ov == bestV && oi < bestI)) { bestV = ov; bestI = oi; }
  }
  if (lane == 0) { sval[wave] = bestV; sidx[wave] = bestI; }
  __syncthreads();

  if (tid == 0) {
    float bv = sval[0];
    int   bi = sidx[0];
#pragma unroll
    for (int w = 1; w < WAVES; ++w)
      if (sval[w] > bv || (sval[w] == bv && sidx[w] < bi)) { bv = sval[w]; bi = sidx[w]; }

    const int x = bi >> 8;        // index // 256
    const int y = bi & 255;       // index %  256

    const float* g  = gt + (size_t)bj * 11;
    const float  gx = g[9];
    const float  gy = g[10];

    float loss = 0.0f;
    if (gx >= 0.0f && gy >= 0.0f && gx < (float)Mdim && gy < (float)Ndim) {
      const float xf = (float)x, yf = (float)y;
      float acc = (gx - xf) * (gx - xf) + (gy - yf) * (gy - yf);   // xy_loss
      acc += (1.0f - bv) * (1.0f - bv);                            // conf_loss
      const float* pp = pred + (size_t)b * Cn * MN + (size_t)(x * Ndim + y);
#pragma unroll
      for (int c = 0; c < Cn; ++c) {                               // class_loss
        const float d = pp[(size_t)c * MN] - g[c];
        acc += d * d;
      }
      loss = acc;
    }
    ws[bj] = loss;
  }
}

// Deterministic per-batch sum over J (fixed order, no atomics).
extern "C" __global__ void labelloss_reduce_kernel(const float* __restrict__ ws,
                                                   float* __restrict__ out) {
  const int b = threadIdx.x;
  if (b < Bn) {
    float s = 0.0f;
#pragma unroll
    for (int j = 0; j < Jn; ++j) s += ws[b * Jn + j];
    out[b] = s;
  }
}

extern "C" void kernel_launch(void* const* d_in, const int* in_sizes, int n_in,
                              void* d_out, int out_size, void* d_ws, size_t ws_size,
                              hipStream_t stream) {
  (void)in_sizes; (void)n_in; (void)out_size; (void)ws_size;
  const float* pred = (const float*)d_in[0];   // (B, C, M, N) f32
  const float* gt   = (const float*)d_in[1];   // (B, J, 11)   f32
  const float* hm   = (const float*)d_in[2];   // (B, J, M, N) f32
  float* ws  = (float*)d_ws;                   // 352 floats of scratch
  float* out = (float*)d_out;                  // 32 floats

  labelloss_argmax_kernel<<<dim3(Bn * Jn), dim3(THREADS), 0, stream>>>(pred, gt, hm, ws);
  labelloss_reduce_kernel<<<dim3(1), dim3(32), 0, stream>>>(ws, out);
}